// GCMC_8461085573265
// MI455X (gfx1250) — compile-verified
//
#include <hip/hip_runtime.h>
#include <math.h>

// ---------------------------------------------------------------------------
// GCMC / LightGCN-style forward on MI455X (gfx1250, wave32)
//   spmm (atomic scatter, L2-resident dense operand) ->
//   dense layer: leakyrelu(X@Wg+bg)@Wm+bm via V_WMMA_F32_16X16X4_F32 ->
//   spmm -> dense -> BPR + reg loss reduction
//
// Round-2 changes:
//  * Weights stored TRANSPOSED in LDS with row stride 68 dwords:
//    B fragment = one aligned ds_load_b64, conflict-free banks, no v_mov glue.
//  * GEMM2 staging tile padded to stride 68: conflict-free A-frag reloads.
//  * global_prefetch of the next edge chunk in the SpMM stream.
// ---------------------------------------------------------------------------

#define U_CNT 100000
#define I_CNT 50000
#define N_CNT 150000          // U + I, multiple of 16
#define D_DIM 64
#define B_CNT 8192
#define NEG_SLOPE 0.2f
#define REG_LAMBDA 0.0001f
#define WPAD 68               // padded LDS row stride (dwords): 4*col+k bank map

typedef float v2f __attribute__((ext_vector_type(2)));
typedef float v8f __attribute__((ext_vector_type(8)));

// ---------------------------------------------------------------------------
// SpMM: out[rows[e], :] += vals[e] * x[cols[e], :]
// One wave per edge (grid-stride). Lane l handles dims {2l, 2l+1}.
// Dense operand (38.4MB) is L2-resident on MI455X (192MB L2); atomics run at L2.
// ---------------------------------------------------------------------------
__global__ __launch_bounds__(256) void spmm_kernel(
    const int* __restrict__ rows, const int* __restrict__ cols,
    const float* __restrict__ vals,
    const float* __restrict__ ue, const float* __restrict__ ie,
    const float* __restrict__ xin, int use_concat,
    float* __restrict__ out, int E)
{
  int lane = threadIdx.x & 31;
  int wid  = (int)((blockIdx.x * blockDim.x + threadIdx.x) >> 5);
  int nw   = (int)((gridDim.x * blockDim.x) >> 5);
  int d    = lane * 2;
  for (int e = wid; e < E; e += nw) {
    // prefetch the next edge record of this wave's stream (global_prefetch_b8)
    int ep = e + nw;
    if (ep < E) {
      __builtin_prefetch(&rows[ep], 0, 3);
      __builtin_prefetch(&cols[ep], 0, 3);
      __builtin_prefetch(&vals[ep], 0, 3);
    }
    int   r = rows[e];
    int   c = cols[e];
    float v = vals[e];
    const float* src;
    if (use_concat) {
      src = (c < U_CNT) ? (ue + (size_t)c * D_DIM)
                        : (ie + (size_t)(c - U_CNT) * D_DIM);
    } else {
      src = xin + (size_t)c * D_DIM;
    }
    v2f xv = *(const v2f*)(src + d);
    float* dst = out + (size_t)r * D_DIM + d;
    atomicAdd(dst + 0, v * xv.x);
    atomicAdd(dst + 1, v * xv.y);
  }
}

// ---------------------------------------------------------------------------
// Dense layer: Y = leakyrelu(X @ Wg + bg) @ Wm + bm          (N x 64, D=64)
// 128 threads = 4 waves; each wave owns a 16-row strip (4 strips / block).
// Both GEMMs use V_WMMA_F32_16X16X4_F32 (fp32 exact).
//   A frag (16x4): lane (l&15)=row M, (l>>4) selects K-pair -> contiguous v2f
//   B frag (4x16): lane (l&15)=col N -> one aligned ds_load_b64 from W^T
//   C/D (16x16):   VGPR j = row j + 8*(l>>4), lane (l&15)=col
// Intermediate (gcn) staged through per-wave padded LDS to re-enter A layout.
// LDS: 2*17KB transposed weights + 17KB staging + 512B biases = 52.7KB / WG.
// ---------------------------------------------------------------------------
__global__ __launch_bounds__(128) void dense_kernel(
    const float* __restrict__ X,
    const float* __restrict__ Wg, const float* __restrict__ bg,
    const float* __restrict__ Wm, const float* __restrict__ bm,
    float* __restrict__ Y, int nrows)
{
  __shared__ float sWgT[D_DIM * WPAD];   // [col][k], padded
  __shared__ float sWmT[D_DIM * WPAD];   // [col][k], padded
  __shared__ float sbg[D_DIM];
  __shared__ float sbm[D_DIM];
  __shared__ float stage[4][16 * WPAD];  // per-wave gcn tile, padded

  // cooperative transposed weight load: W[k][j] -> W^T[j][k]
  for (int i = threadIdx.x; i < D_DIM * D_DIM; i += 128) {
    int k = i >> 6;          // row of W
    int j = i & 63;          // col of W
    sWgT[j * WPAD + k] = Wg[i];
    sWmT[j * WPAD + k] = Wm[i];
  }
  if (threadIdx.x < D_DIM) {
    sbg[threadIdx.x] = bg[threadIdx.x];
    sbm[threadIdx.x] = bm[threadIdx.x];
  }
  __syncthreads();

  const int wave  = threadIdx.x >> 5;
  const int lane  = threadIdx.x & 31;
  const int row0  = (int)blockIdx.x * 64 + wave * 16;
  if (row0 >= nrows) return;                 // strip-granular: EXEC stays all-1s

  const int mrow  = lane & 15;               // row (A) / col (B,C/D) within tile
  const int khalf = (lane >> 4) * 2;         // K sub-pair select

  // ---- GEMM1: A fragments from global X (contiguous 8B loads) ----
  v2f a[16];
  const float* xrow = X + (size_t)(row0 + mrow) * D_DIM;
  #pragma unroll
  for (int kt = 0; kt < 16; ++kt)
    a[kt] = *(const v2f*)(xrow + kt * 4 + khalf);

  v8f zero = {};
  v8f acc[4] = {zero, zero, zero, zero};

  #pragma unroll
  for (int kt = 0; kt < 16; ++kt) {
    const int krow = kt * 4 + khalf;
    #pragma unroll
    for (int nt = 0; nt < 4; ++nt) {
      const int col = nt * 16 + mrow;
      v2f b = *(const v2f*)(sWgT + col * WPAD + krow);   // one ds_load_b64
      acc[nt] = __builtin_amdgcn_wmma_f32_16x16x4_f32(
          false, a[kt], false, b, (short)0, acc[nt], false, false);
    }
  }

  // ---- bias + leaky ReLU, stage gcn tile (row-major, padded) in LDS ----
  float* st = stage[wave];
  #pragma unroll
  for (int nt = 0; nt < 4; ++nt) {
    const int col = nt * 16 + mrow;
    const float bb = sbg[col];
    #pragma unroll
    for (int j = 0; j < 8; ++j) {
      const int M = j + 8 * (lane >> 4);
      float x = acc[nt][j] + bb;
      x = (x > 0.0f) ? x : NEG_SLOPE * x;
      st[M * WPAD + col] = x;
    }
  }
  // same-wave LDS RAW: hardware DS ordering + compiler dscnt waits suffice.

  // ---- GEMM2: A fragments from staged gcn (conflict-free), B from sWmT ----
  v2f a2[16];
  #pragma unroll
  for (int kt = 0; kt < 16; ++kt)
    a2[kt] = *(const v2f*)(st + mrow * WPAD + kt * 4 + khalf);

  v8f acc2[4] = {zero, zero, zero, zero};
  #pragma unroll
  for (int kt = 0; kt < 16; ++kt) {
    const int krow = kt * 4 + khalf;
    #pragma unroll
    for (int nt = 0; nt < 4; ++nt) {
      const int col = nt * 16 + mrow;
      v2f b = *(const v2f*)(sWmT + col * WPAD + krow);   // one ds_load_b64
      acc2[nt] = __builtin_amdgcn_wmma_f32_16x16x4_f32(
          false, a2[kt], false, b, (short)0, acc2[nt], false, false);
    }
  }

  #pragma unroll
  for (int nt = 0; nt < 4; ++nt) {
    const int col = nt * 16 + mrow;
    const float bb = sbm[col];
    #pragma unroll
    for (int j = 0; j < 8; ++j) {
      const int M = j + 8 * (lane >> 4);
      Y[(size_t)(row0 + M) * D_DIM + col] = acc2[nt][j] + bb;
    }
  }
}

// ---------------------------------------------------------------------------
// BPR + reg loss. One wave32 per batch element; norms of layer-1/2 rows are
// computed on the fly (only 3*8192 rows ever need normalizing).
// ---------------------------------------------------------------------------
__device__ __forceinline__ float wred(float v) {
  #pragma unroll
  for (int m = 16; m > 0; m >>= 1) v += __shfl_xor(v, m, 32);
  return v;
}

__global__ __launch_bounds__(256) void loss_kernel(
    const int* __restrict__ user, const int* __restrict__ posi,
    const int* __restrict__ nega,
    const float* __restrict__ ue, const float* __restrict__ ie,
    const float* __restrict__ ego1, const float* __restrict__ ego2,
    float* __restrict__ acc)
{
  int lane = threadIdx.x & 31;
  int i    = (int)((blockIdx.x * blockDim.x + threadIdx.x) >> 5);
  if (i >= B_CNT) return;
  int uu = user[i], pp = posi[i], nn = nega[i];
  int d  = lane * 2;

  v2f eu = *(const v2f*)(ue + (size_t)uu * D_DIM + d);
  v2f ep = *(const v2f*)(ie + (size_t)pp * D_DIM + d);
  v2f en = *(const v2f*)(ie + (size_t)nn * D_DIM + d);
  v2f u1 = *(const v2f*)(ego1 + (size_t)uu * D_DIM + d);
  v2f p1 = *(const v2f*)(ego1 + (size_t)(U_CNT + pp) * D_DIM + d);
  v2f n1 = *(const v2f*)(ego1 + (size_t)(U_CNT + nn) * D_DIM + d);
  v2f u2 = *(const v2f*)(ego2 + (size_t)uu * D_DIM + d);
  v2f p2 = *(const v2f*)(ego2 + (size_t)(U_CNT + pp) * D_DIM + d);
  v2f n2 = *(const v2f*)(ego2 + (size_t)(U_CNT + nn) * D_DIM + d);

  float pos0 = wred(eu.x * ep.x + eu.y * ep.y);
  float neg0 = wred(eu.x * en.x + eu.y * en.y);
  float squ  = wred(eu.x * eu.x + eu.y * eu.y);
  float sqp  = wred(ep.x * ep.x + ep.y * ep.y);
  float sqn  = wred(en.x * en.x + en.y * en.y);
  float s1u  = wred(u1.x * u1.x + u1.y * u1.y);
  float s1p  = wred(p1.x * p1.x + p1.y * p1.y);
  float s1n  = wred(n1.x * n1.x + n1.y * n1.y);
  float d1p  = wred(u1.x * p1.x + u1.y * p1.y);
  float d1n  = wred(u1.x * n1.x + u1.y * n1.y);
  float s2u  = wred(u2.x * u2.x + u2.y * u2.y);
  float s2p  = wred(p2.x * p2.x + p2.y * p2.y);
  float s2n  = wred(n2.x * n2.x + n2.y * n2.y);
  float d2p  = wred(u2.x * p2.x + u2.y * p2.y);
  float d2n  = wred(u2.x * n2.x + u2.y * n2.y);

  if (lane == 0) {
    float nu1 = fmaxf(sqrtf(s1u), 1e-12f);
    float np1 = fmaxf(sqrtf(s1p), 1e-12f);
    float nn1 = fmaxf(sqrtf(s1n), 1e-12f);
    float nu2 = fmaxf(sqrtf(s2u), 1e-12f);
    float np2 = fmaxf(sqrtf(s2p), 1e-12f);
    float nn2 = fmaxf(sqrtf(s2n), 1e-12f);
    float pos_s = pos0 + d1p / (nu1 * np1) + d2p / (nu2 * np2);
    float neg_s = neg0 + d1n / (nu1 * nn1) + d2n / (nu2 * nn2);
    float t  = neg_s - pos_s;
    float sp = fmaxf(t, 0.0f) + log1pf(expf(-fabsf(t)));   // stable softplus
    atomicAdd(acc + 0, sp);
    atomicAdd(acc + 1, squ + sqp + sqn);
  }
}

__global__ void finalize_kernel(const float* __restrict__ acc,
                                float* __restrict__ out)
{
  if (threadIdx.x == 0 && blockIdx.x == 0) {
    out[0] = acc[0] / (float)B_CNT;
    out[1] = REG_LAMBDA * 0.5f * acc[1] / (float)B_CNT;
  }
}

// ---------------------------------------------------------------------------
extern "C" void kernel_launch(void* const* d_in, const int* in_sizes, int n_in,
                              void* d_out, int out_size, void* d_ws, size_t ws_size,
                              hipStream_t stream)
{
  const int*   user     = (const int*)d_in[0];
  const int*   positive = (const int*)d_in[1];
  const int*   negative = (const int*)d_in[2];
  const int*   rows     = (const int*)d_in[3];
  const int*   cols     = (const int*)d_in[4];
  const float* vals     = (const float*)d_in[5];
  const float* user_emb = (const float*)d_in[6];
  const float* item_emb = (const float*)d_in[7];
  const float* Wg0 = (const float*)d_in[8];
  const float* bg0 = (const float*)d_in[9];
  const float* Wm0 = (const float*)d_in[10];
  const float* bm0 = (const float*)d_in[11];
  const float* Wg1 = (const float*)d_in[12];
  const float* bg1 = (const float*)d_in[13];
  const float* Wm1 = (const float*)d_in[14];
  const float* bm1 = (const float*)d_in[15];
  const int E = in_sizes[3];

  const size_t mat = (size_t)N_CNT * D_DIM;
  float* side = (float*)d_ws;
  float* ego1 = side + mat;
  float* ego2 = ego1 + mat;
  float* acc  = ego2 + mat;

  const int dense_grid = (N_CNT + 63) / 64;
  const int loss_grid  = (B_CNT * 32) / 256;

  // layer 1
  hipMemsetAsync(side, 0, mat * sizeof(float), stream);
  hipMemsetAsync(acc, 0, 2 * sizeof(float), stream);
  spmm_kernel<<<2048, 256, 0, stream>>>(rows, cols, vals, user_emb, item_emb,
                                        nullptr, 1, side, E);
  dense_kernel<<<dense_grid, 128, 0, stream>>>(side, Wg0, bg0, Wm0, bm0,
                                               ego1, N_CNT);
  // layer 2
  hipMemsetAsync(side, 0, mat * sizeof(float), stream);
  spmm_kernel<<<2048, 256, 0, stream>>>(rows, cols, vals, user_emb, item_emb,
                                        ego1, 0, side, E);
  dense_kernel<<<dense_grid, 128, 0, stream>>>(side, Wg1, bg1, Wm1, bm1,
                                               ego2, N_CNT);
  // loss
  loss_kernel<<<loss_grid, 256, 0, stream>>>(user, positive, negative,
                                             user_emb, item_emb, ego1, ego2, acc);
  finalize_kernel<<<1, 1, 0, stream>>>(acc, (float*)d_out);
}